// CrossAttention_27908697489923
// MI455X (gfx1250) — compile-verified
//
#include <hip/hip_runtime.h>
#include <math.h>

// ---------------- problem constants (from reference) ----------------
#define BB   2
#define NN   4096
#define DD   320
#define HH   8
#define DH   40
#define NH   2048            // N/2 == merged length m (since r == N/2)
#define SCALE 0.15811388300841898f   // 40^-0.5

typedef float cv2f __attribute__((ext_vector_type(2)));
typedef float cv8f __attribute__((ext_vector_type(8)));
typedef unsigned int cv4u __attribute__((ext_vector_type(4)));
typedef int cv8i __attribute__((ext_vector_type(8)));
typedef int cv4i __attribute__((ext_vector_type(4)));

// D = A(16x4) * B(4x16) + C, full fp32 matrix op (CDNA5 WMMA)
static __device__ __forceinline__ cv8f wmma4(cv2f a, cv2f b, cv8f c) {
    return __builtin_amdgcn_wmma_f32_16x16x4_f32(
        /*neg_a=*/false, a, /*neg_b=*/false, b,
        /*c_mod=*/(short)0, c, /*reuse_a=*/false, /*reuse_b=*/false);
}

// ---------------------------------------------------------------
// TDM: DMA a 2D fp32 tile (width elts x height rows, row stride in elts)
// from global memory into LDS at byte offset lds_off. D# built per
// CDNA5 ISA ch.8 (group0: count/lds/global/type, group1: dims+strides).
// Tracked on TENSORcnt. clang-23 toolchain: 6-arg builtin form.
// ---------------------------------------------------------------
static __device__ __forceinline__ void tdm_load_2d(const void* gptr,
                                                   unsigned lds_off,
                                                   unsigned width_elts,
                                                   unsigned height,
                                                   unsigned row_stride_elts) {
    const unsigned long long ga = (unsigned long long)(uintptr_t)gptr;
    cv4u g0;
    g0.x = 1u;                                   // count=1, user descriptor
    g0.y = lds_off;                              // lds_addr (bytes)
    g0.z = (unsigned)(ga & 0xFFFFFFFFu);         // global_addr[31:0]
    g0.w = (unsigned)((ga >> 32) & 0x01FFFFFFu)  // global_addr[56:32]
         | (2u << 30);                           // type = 2 ("image")
    cv8i g1;
    g1[0] = (int)(2u << 16);                               // data_size=4B
    g1[1] = (int)((width_elts & 0xFFFFu) << 16);           // tensor_dim0 lo
    g1[2] = (int)((width_elts >> 16) |
                  ((height & 0xFFFFu) << 16));             // dim0 hi | dim1 lo
    g1[3] = (int)((height >> 16) |
                  ((width_elts & 0xFFFFu) << 16));         // dim1 hi | tile_dim0
    g1[4] = (int)(height & 0xFFFFu);                       // tile_dim1 (tile_dim2=0)
    g1[5] = (int)row_stride_elts;                          // tensor_dim0_stride lo32
    g1[6] = 0;                                             // stride hi | dim1_stride lo
    g1[7] = 0;
    const cv4i z4 = {0, 0, 0, 0};
    const cv8i z8 = {0, 0, 0, 0, 0, 0, 0, 0};
    __builtin_amdgcn_tensor_load_to_lds(g0, g1, z4, z4, z8, 0);
}

// ---------------------------------------------------------------
// Generic fp32 GEMM: C[M x Nn] = X[M x K] * W[K x Nn] (+ bias)
// grid.x = M/16, grid.y = Nn/64, one wave per block, 16x64 per wave.
// ---------------------------------------------------------------
__global__ __launch_bounds__(32)
void gemm_wmma(const float* __restrict__ X, const float* __restrict__ W,
               const float* __restrict__ bias, float* __restrict__ C,
               int M, int K, int Nn) {
    const int m0   = blockIdx.x * 16;
    const int n0   = blockIdx.y * 64;
    const int lane = threadIdx.x;
    const int half = lane >> 4;
    const int r    = lane & 15;

    cv8f acc[4] = {};

    for (int k = 0; k < K; k += 4) {
        // A fragment: lane r holds row m0+r, cols k+2*half, k+2*half+1
        cv2f a;
        const float* ap = X + (size_t)(m0 + r) * K + k + 2 * half;
        a.x = ap[0]; a.y = ap[1];
#pragma unroll
        for (int t = 0; t < 4; ++t) {
            // B fragment: rows k+2*half, k+2*half+1 of W, col n0+16t+r
            cv2f b;
            const float* bp = W + (size_t)(k + 2 * half) * Nn + n0 + 16 * t + r;
            b.x = bp[0];
            b.y = bp[Nn];
            acc[t] = wmma4(a, b, acc[t]);
        }
    }
#pragma unroll
    for (int t = 0; t < 4; ++t) {
#pragma unroll
        for (int g = 0; g < 8; ++g) {
            const int row = m0 + g + 8 * half;
            const int col = n0 + 16 * t + r;
            float vv = acc[t][g];
            if (bias) vv += bias[col];
            C[(size_t)row * Nn + col] = vv;
        }
    }
}

// ---------------------------------------------------------------
// k_mean over heads + L2 normalize, split even->am / odd->bm.
// One wave per (b,n) row. grid.x = B*N.
// ---------------------------------------------------------------
__global__ __launch_bounds__(32)
void kmean_norm(const float* __restrict__ k, float* __restrict__ am,
                float* __restrict__ bm) {
    const int bn   = blockIdx.x;
    const int lane = threadIdx.x;
    const float* kp = k + (size_t)bn * DD;

    float sx = 0.f, sy = 0.f;
    if (lane < 20) {
#pragma unroll
        for (int h = 0; h < HH; ++h) {
            sx += kp[h * DH + 2 * lane];
            sy += kp[h * DH + 2 * lane + 1];
        }
        sx *= 0.125f; sy *= 0.125f;
    }
    float ss = sx * sx + sy * sy;
#pragma unroll
    for (int m = 16; m >= 1; m >>= 1) ss += __shfl_xor(ss, m, 32);
    const float inv = 1.0f / sqrtf(ss);

    const int b = bn / NN;
    const int n = bn % NN;
    float* outp = ((n & 1) ? bm : am) + ((size_t)b * NH + (n >> 1)) * DH;
    if (lane < 20) {
        outp[2 * lane]     = sx * inv;
        outp[2 * lane + 1] = sy * inv;
    }
}

// ---------------------------------------------------------------
// scores = am @ bm^T per batch (2048x2048, K=40) via WMMA,
// keep running row argmax only. grid.x = B * 128 src tiles.
// ---------------------------------------------------------------
__global__ __launch_bounds__(32)
void merge_argmax(const float* __restrict__ am, const float* __restrict__ bm,
                  int* __restrict__ node_idx) {
    const int b    = blockIdx.x >> 7;
    const int it   = blockIdx.x & 127;
    const int lane = threadIdx.x;
    const int half = lane >> 4;
    const int r    = lane & 15;

    const float* A  = am + (size_t)b * NH * DH;
    const float* Bm = bm + (size_t)b * NH * DH;

    cv2f afr[10];
#pragma unroll
    for (int s = 0; s < 10; ++s) {
        const float* ap = A + (size_t)(it * 16 + r) * DH + 4 * s + 2 * half;
        afr[s].x = ap[0]; afr[s].y = ap[1];
    }

    float vmax[8]; int vidx[8];
#pragma unroll
    for (int g = 0; g < 8; ++g) { vmax[g] = -3.0e38f; vidx[g] = 0; }

    for (int jt = 0; jt < 128; ++jt) {
        cv8f c = {};
#pragma unroll
        for (int s = 0; s < 10; ++s) {
            // Bmat[k][j] = bm[j][k] -> contiguous float2 per lane
            cv2f bfr;
            const float* bp = Bm + (size_t)(jt * 16 + r) * DH + 4 * s + 2 * half;
            bfr.x = bp[0]; bfr.y = bp[1];
            c = wmma4(afr[s], bfr, c);
        }
        const int j = jt * 16 + r;
#pragma unroll
        for (int g = 0; g < 8; ++g) {
            const float sv = c[g];
            if (sv > vmax[g] || (sv == vmax[g] && j < vidx[g])) {
                vmax[g] = sv; vidx[g] = j;
            }
        }
    }
    // reduce across the 16 column lanes of each half (first occurrence on tie)
#pragma unroll
    for (int g = 0; g < 8; ++g) {
        float mv = vmax[g]; int mi = vidx[g];
#pragma unroll
        for (int d = 1; d < 16; d <<= 1) {
            const float ov = __shfl_xor(mv, d, 32);
            const int   oi = __shfl_xor(mi, d, 32);
            if (ov > mv || (ov == mv && oi < mi)) { mv = ov; mi = oi; }
        }
        vmax[g] = mv; vidx[g] = mi;
    }
    if (r == 0) {
#pragma unroll
        for (int g = 0; g < 8; ++g)
            node_idx[(size_t)b * NH + it * 16 + 8 * half + g] = vidx[g];
    }
}

// ---------------------------------------------------------------
// Deterministic merge: one wave per (b, dst j). Gather all src rows
// whose argmax == j (instead of float atomics -> replay-deterministic).
// grid.x = B * 2048.
// ---------------------------------------------------------------
__global__ __launch_bounds__(32)
void merge_gather(const float* __restrict__ k, const float* __restrict__ v,
                  const int* __restrict__ node_idx,
                  float* __restrict__ km, float* __restrict__ vm) {
    const int b    = blockIdx.x >> 11;
    const int j    = blockIdx.x & (NH - 1);
    const int lane = threadIdx.x;

    const float* kb  = k + (size_t)b * NN * DD;
    const float* vb  = v + (size_t)b * NN * DD;
    const int*   idx = node_idx + (size_t)b * NH;

    float ks[10], vs[10];
#pragma unroll
    for (int c = 0; c < 10; ++c) {
        ks[c] = kb[(size_t)(2 * j + 1) * DD + c * 32 + lane];
        vs[c] = vb[(size_t)(2 * j + 1) * DD + c * 32 + lane];
    }
    float cnt = 1.0f;

    for (int i0 = 0; i0 < NH; i0 += 32) {
        const int myidx = idx[i0 + lane];
#pragma unroll 4
        for (int t = 0; t < 32; ++t) {
            const int jj = __shfl(myidx, t, 32);   // wave-uniform
            if (jj == j) {
                const float* kr = kb + (size_t)(2 * (i0 + t)) * DD;
                const float* vr = vb + (size_t)(2 * (i0 + t)) * DD;
#pragma unroll
                for (int c = 0; c < 10; ++c) {
                    ks[c] += kr[c * 32 + lane];
                    vs[c] += vr[c * 32 + lane];
                }
                cnt += 1.0f;
            }
        }
    }
    const float inv = 1.0f / cnt;
#pragma unroll
    for (int c = 0; c < 10; ++c) {
        km[((size_t)b * NH + j) * DD + c * 32 + lane] = ks[c] * inv;
        vm[((size_t)b * NH + j) * DD + c * 32 + lane] = vs[c] * inv;
    }
}

// ---------------------------------------------------------------
// Flash attention, TDM-staged K/V:
//   block = 256 threads (8 waves = 8 query tiles of 16), grid (N/128, H, B).
//   Wave 0 double-buffers the 16x40 fp32 K and V tiles into LDS with
//   tensor_load_to_lds (TENSORcnt), all waves consume from LDS.
//   S = Q Kt (10 WMMA) -> online softmax -> P staged via private LDS
//   (s_wait_dscnt) -> O += P V (12 WMMA, DH padded to 48).
// ---------------------------------------------------------------
__global__ __launch_bounds__(256)
void flash_attn(const float* __restrict__ q, const float* __restrict__ km,
                const float* __restrict__ vm, float* __restrict__ out) {
    __shared__ float ldsK[2][16 * DH];
    __shared__ float ldsV[2][16 * DH];
    __shared__ float pbuf[8][16 * 16];

    const int h = blockIdx.y, b = blockIdx.z;
    const int w    = threadIdx.x >> 5;
    const int lane = threadIdx.x & 31;
    const int half = lane >> 4;
    const int r    = lane & 15;
    const int qrow0 = blockIdx.x * 128 + w * 16;

    const float* qb = q  + ((size_t)b * NN + qrow0) * DD + h * DH;
    const float* kb = km + (size_t)b * NH * DD + h * DH;
    const float* vb = vm + (size_t)b * NH * DD + h * DH;

    cv2f qfr[10];
#pragma unroll
    for (int s = 0; s < 10; ++s) {
        const float* qp = qb + (size_t)r * DD + 4 * s + 2 * half;
        qfr[s].x = qp[0]; qfr[s].y = qp[1];
    }

    float* myp = pbuf[w];
    const bool tw = (threadIdx.x < 32);   // wave 0 drives the TDM

    if (tw) {   // prologue: prefetch tiles 0 and 1
        tdm_load_2d(kb,           (unsigned)(uintptr_t)&ldsK[0][0], DH, 16, DD);
        tdm_load_2d(vb,           (unsigned)(uintptr_t)&ldsV[0][0], DH, 16, DD);
        tdm_load_2d(kb + 16 * DD, (unsigned)(uintptr_t)&ldsK[1][0], DH, 16, DD);
        tdm_load_2d(vb + 16 * DD, (unsigned)(uintptr_t)&ldsV[1][0], DH, 16, DD);
    }

    cv8f o0 = {}, o1 = {}, o2 = {};
    float mrow[8], lrow[8];
#pragma unroll
    for (int g = 0; g < 8; ++g) { mrow[g] = -3.0e38f; lrow[g] = 0.f; }

    for (int jt = 0; jt < NH / 16; ++jt) {
        const int cur = jt & 1;
        if (tw) {
            // in-order TENSORcnt: <=2 outstanding => pair jt has landed
            if (jt < NH / 16 - 2) __builtin_amdgcn_s_wait_tensorcnt(2);
            else                  __builtin_amdgcn_s_wait_tensorcnt(0);
        }
        __syncthreads();                       // tile jt visible to all waves

        const float* Kt = ldsK[cur];
        const float* Vt = ldsV[cur];

        // ---- S = Q Kt ----
        cv8f sc = {};
#pragma unroll
        for (int s = 0; s < 10; ++s) {
            cv2f bfr;                           // Kt[c][j] = K[j][c]
            const float* bp = Kt + r * DH + 4 * s + 2 * half;
            bfr.x = bp[0]; bfr.y = bp[1];
            sc = wmma4(qfr[s], bfr, sc);
        }

        // ---- online softmax ----
        float p[8], alpha[8];
#pragma unroll
        for (int g = 0; g < 8; ++g) {
            const float sv = sc[g] * SCALE;
            float tm = sv;
#pragma unroll
            for (int d = 1; d < 16; d <<= 1) tm = fmaxf(tm, __shfl_xor(tm, d, 32));
            const float nm = fmaxf(mrow[g], tm);
            alpha[g] = __expf(mrow[g] - nm);
            p[g]     = __expf(sv - nm);
            float rs = p[g];
#pragma unroll
            for (int d = 1; d < 16; d <<= 1) rs += __shfl_xor(rs, d, 32);
            lrow[g] = lrow[g] * alpha[g] + rs;
            mrow[g] = nm;
        }

        // ---- stage P (C-layout -> A-layout) via this wave's LDS region ----
#pragma unroll
        for (int g = 0; g < 8; ++g) myp[(g + 8 * half) * 16 + r] = p[g];
        asm volatile("s_wait_dscnt 0x0" ::: "memory");   // wave-local ordering

#pragma unroll
        for (int g = 0; g < 8; ++g) {
            o0[g] *= alpha[g]; o1[g] *= alpha[g]; o2[g] *= alpha[g];
        }

        // ---- O += P V ----
#pragma unroll
        for (int s = 0; s < 4; ++s) {
            cv2f pa;
            const float* pp = myp + r * 16 + 4 * s + 2 * half;
            pa.x = pp[0]; pa.y = pp[1];

            const float* vrow = Vt + (4 * s + 2 * half) * DH;
            cv2f b0, b1, b2;
            b0.x = vrow[r];           b0.y = vrow[DH + r];
            b1.x = vrow[16 + r];      b1.y = vrow[DH + 16 + r];
            const bool okc = (32 + r) < DH;          // pad cols 40..47 with 0
            b2.x = okc ? vrow[32 + r]      : 0.f;
            b2.y = okc ? vrow[DH + 32 + r] : 0.f;

            o0 = wmma4(pa, b0, o0);
            o1 = wmma4(pa, b1, o1);
            o2 = wmma4(pa, b2, o2);
        }

        __syncthreads();                       // everyone done with buffer cur
        if (tw && jt + 2 < NH / 16) {          // refill cur with tile jt+2
            const size_t off = (size_t)(jt + 2) * 16 * DD;
            tdm_load_2d(kb + off, (unsigned)(uintptr_t)&ldsK[cur][0], DH, 16, DD);
            tdm_load_2d(vb + off, (unsigned)(uintptr_t)&ldsV[cur][0], DH, 16, DD);
        }
    }

    // epilogue: divide by l, store to out[b, row, h*40 + col]
#pragma unroll
    for (int g = 0; g < 8; ++g) {
        const float linv = 1.0f / lrow[g];
        const size_t row = (size_t)b * NN + qrow0 + g + 8 * half;
        float* op = out + row * DD + h * DH;
        op[r]      = o0[g] * linv;
        op[16 + r] = o1[g] * linv;
        if (32 + r < DH) op[32 + r] = o2[g] * linv;
    }
}

// ---------------------------------------------------------------
extern "C" void kernel_launch(void* const* d_in, const int* in_sizes, int n_in,
                              void* d_out, int out_size, void* d_ws, size_t ws_size,
                              hipStream_t stream) {
    (void)in_sizes; (void)n_in; (void)out_size; (void)ws_size;
    const float* x    = (const float*)d_in[0];
    const float* Wq   = (const float*)d_in[1];
    const float* Wk   = (const float*)d_in[2];
    const float* Wv   = (const float*)d_in[3];
    const float* Wout = (const float*)d_in[4];
    const float* bout = (const float*)d_in[5];
    // d_in[6] = r (always N/2 = 2048 for this problem; merge logic specialized)

    float* ws = (float*)d_ws;
    const size_t QS = (size_t)BB * NN * DD;   // 2,621,440
    const size_t MS = (size_t)BB * NH * DD;   // 1,310,720
    const size_t AS = (size_t)BB * NH * DH;   //   163,840

    float* q_    = ws;
    float* k_    = ws + QS;
    float* v_    = ws + 2 * QS;
    float* attn  = ws + 3 * QS;
    float* kmrg  = ws + 4 * QS;
    float* vmrg  = ws + 4 * QS + MS;
    float* am    = ws + 4 * QS + 2 * MS;
    float* bm    = am + AS;
    int*   nidx  = (int*)(bm + AS);

    const dim3 gg((BB * NN) / 16, DD / 64, 1);   // (512, 5)

    // 1) QKV projections (fp32 WMMA GEMMs)
    gemm_wmma<<<gg, 32, 0, stream>>>(x, Wq, nullptr, q_, BB * NN, DD, DD);
    gemm_wmma<<<gg, 32, 0, stream>>>(x, Wk, nullptr, k_, BB * NN, DD, DD);
    gemm_wmma<<<gg, 32, 0, stream>>>(x, Wv, nullptr, v_, BB * NN, DD, DD);

    // 2) metric = normalize(mean over heads of k), split even/odd
    kmean_norm<<<BB * NN, 32, 0, stream>>>(k_, am, bm);

    // 3) per-src argmax over 2048 dst scores (WMMA)
    merge_argmax<<<BB * 128, 32, 0, stream>>>(am, bm, nidx);

    // 4) deterministic scatter-mean merge of k and v
    merge_gather<<<BB * NH, 32, 0, stream>>>(k_, v_, nidx, kmrg, vmrg);

    // 5) flash attention over merged k/v (WMMA + TDM double buffering)
    flash_attn<<<dim3(NN / 128, HH, BB), 256, 0, stream>>>(q_, kmrg, vmrg, attn);

    // 6) output projection + bias into d_out
    gemm_wmma<<<gg, 32, 0, stream>>>(attn, Wout, bout, (float*)d_out,
                                     BB * NN, DD, DD);
}